// Sample_Group_446676598875
// MI455X (gfx1250) — compile-verified
//
#include <hip/hip_runtime.h>
#include <math.h>

#define B_  4
#define N_  16384
#define D_  62
#define CH  64
#define S_  2048
#define NS  64
// f32(0.1**2 computed in float64) == f32(0.01) -> use 0.01f, NOT 0.1f*0.1f (1 ulp off)
#define R2  0.01f

typedef float v2f __attribute__((ext_vector_type(2)));
typedef float v8f __attribute__((ext_vector_type(8)));

// ---------------- prep kernels ----------------

__global__ void k_prep_p(const float* __restrict__ xy, float4* __restrict__ pprep) {
    int t = blockIdx.x * blockDim.x + threadIdx.x;
    if (t >= B_ * N_) return;
    float x = xy[t * 2], y = xy[t * 2 + 1];
    pprep[t] = make_float4(-2.0f * x, -2.0f * y, 1.0f, x * x + y * y);
}

__global__ void k_prep_q(const float* __restrict__ xy, const int* __restrict__ fps,
                         float4* __restrict__ qprep, float* __restrict__ newxy) {
    int t = blockIdx.x * blockDim.x + threadIdx.x;
    if (t >= B_ * S_) return;
    int b = t / S_;
    int i = fps[t];
    float x = xy[(b * N_ + i) * 2], y = xy[(b * N_ + i) * 2 + 1];
    newxy[t * 2] = x;
    newxy[t * 2 + 1] = y;
    qprep[t] = make_float4(x, y, x * x + y * y, 1.0f);
}

// ---------------- ball-query via WMMA + ballot bit-scan ----------------
// One wave handles 16 queries. Per 16-point tile one v_wmma_f32_16x16x4_f32:
//   A(16x4) rows = [qx, qy, |q|^2, 1] ; B(4x16) cols = [-2px, -2py, 1, |p|^2]
//   => D[m][n] = squared distance.
// D layout: VGPR v: lanes 0-15 = (row v, col lane); lanes 16-31 = (row v+8, col lane-16).
// So ballot(d[v] <= R2) packs: bits[15:0] = row v hits, bits[31:16] = row v+8 hits.
// Lane r (0..15) owns query row r; selects its mask from the 8 uniform ballots and
// drains set bits (ascending column == ascending index) into its private LDS list.

__global__ __launch_bounds__(32) void k_search(const float4* __restrict__ pprep,
                                               const float4* __restrict__ qprep,
                                               int* __restrict__ idxws) {
    __shared__ int lst[16 * NS];   // per-row private lists (no cross-lane sharing)

    const int lane = threadIdx.x;
    const int b  = blockIdx.x / (S_ / 16);
    const int qt = blockIdx.x % (S_ / 16);
    const int r  = lane & 15;

    float4 q = qprep[b * S_ + qt * 16 + r];
    v2f av;
    if (lane < 16) { av.x = q.x; av.y = q.y; }   // K0=qx, K1=qy
    else           { av.x = q.z; av.y = q.w; }   // K2=|q|^2, K3=1

    int cnt = (lane < 16) ? 0 : NS;              // upper lanes never block early-exit
    const float4* pp = pprep + b * N_;
    const int idxsel = r & 7;

    for (int t = 0; t < N_ / 16; ++t) {
        float4 p = pp[t * 16 + r];
        __builtin_prefetch(&pp[(t + 8) * 16 + r], 0, 3);   // speculative, drop-safe
        v2f bv;
        if (lane < 16) { bv.x = p.x; bv.y = p.y; }   // K0=-2px, K1=-2py
        else           { bv.x = p.z; bv.y = p.w; }   // K2=1,    K3=|p|^2

        v8f cz = {0.f, 0.f, 0.f, 0.f, 0.f, 0.f, 0.f, 0.f};
        v8f d = __builtin_amdgcn_wmma_f32_16x16x4_f32(
            false, av, false, bv, (short)0, cz, false, false);

        // 8 wave-uniform hit masks (rows v and v+8 packed per ballot)
        unsigned bal0 = (unsigned)__ballot(d[0] <= R2);
        unsigned bal1 = (unsigned)__ballot(d[1] <= R2);
        unsigned bal2 = (unsigned)__ballot(d[2] <= R2);
        unsigned bal3 = (unsigned)__ballot(d[3] <= R2);
        unsigned bal4 = (unsigned)__ballot(d[4] <= R2);
        unsigned bal5 = (unsigned)__ballot(d[5] <= R2);
        unsigned bal6 = (unsigned)__ballot(d[6] <= R2);
        unsigned bal7 = (unsigned)__ballot(d[7] <= R2);

        // lane r picks ballot[r & 7], then low half (r<8) or high half (r>=8)
        unsigned t0 = (idxsel & 1) ? bal1 : bal0;
        unsigned t1 = (idxsel & 1) ? bal3 : bal2;
        unsigned t2 = (idxsel & 1) ? bal5 : bal4;
        unsigned t3 = (idxsel & 1) ? bal7 : bal6;
        unsigned u0 = (idxsel & 2) ? t1 : t0;
        unsigned u1 = (idxsel & 2) ? t3 : t2;
        unsigned mf = (idxsel & 4) ? u1 : u0;
        unsigned m  = (r < 8) ? (mf & 0xFFFFu) : (mf >> 16);
        if (lane >= 16) m = 0;

        const int base = t * 16;
        while (m && cnt < NS) {            // ~0-2 iterations typical
            int j = __builtin_ctz(m);
            m &= m - 1;
            lst[r * NS + cnt] = base + j;  // ascending j == ascending global index
            ++cnt;
        }

        if (__all(cnt >= NS)) break;       // uniform early exit (~6x work reduction)
    }

    if (lane < 16) {
        int c0 = (cnt > 0) ? lst[lane * NS] : 0;   // self-hit guarantees cnt>=1
        int* o = idxws + (b * S_ + qt * 16 + lane) * NS;
        for (int k = 0; k < NS; ++k)
            o[k] = (k < cnt) ? lst[lane * NS + k] : c0;
    }
}

// ---------------- per-(b,s) mean + centered sum-of-squares ----------------

__global__ __launch_bounds__(64) void k_stats(const float* __restrict__ xy,
                                              const float* __restrict__ events,
                                              const int* __restrict__ idxws,
                                              float* __restrict__ means,
                                              double* __restrict__ partials) {
    __shared__ int    sidx[NS];
    __shared__ double red[64];
    const int tid = threadIdx.x;
    const int b = blockIdx.x / S_;
    const int s = blockIdx.x % S_;
    sidx[tid] = idxws[(b * S_ + s) * NS + tid];
    __syncthreads();

    const int c = tid;
    double sum = 0.0, ssq = 0.0;
    for (int k = 0; k < NS; ++k) {
        int i = sidx[k];
        float v = (c < D_) ? events[(b * N_ + i) * D_ + c]
                           : xy[(b * N_ + i) * 2 + (c - D_)];
        sum += (double)v;
        ssq += (double)v * (double)v;
    }
    means[(b * S_ + s) * CH + c] = (float)(sum * (1.0 / 64.0));
    red[tid] = ssq - sum * sum * (1.0 / 64.0);   // sum((v-mean)^2), fp64
    __syncthreads();
    for (int off = 32; off > 0; off >>= 1) {
        if (tid < off) red[tid] += red[tid + off];
        __syncthreads();
    }
    if (tid == 0) partials[b * S_ + s] = red[0];
}

// ---------------- per-batch std -> scale ----------------

__global__ __launch_bounds__(256) void k_reduce(const double* __restrict__ partials,
                                                float* __restrict__ scale) {
    __shared__ double red[256];
    const int b = blockIdx.x;
    const int tid = threadIdx.x;
    double acc = 0.0;
    for (int j = tid; j < S_; j += 256) acc += partials[b * S_ + j];  // fixed order
    red[tid] = acc;
    __syncthreads();
    for (int off = 128; off > 0; off >>= 1) {
        if (tid < off) red[tid] += red[tid + off];
        __syncthreads();
    }
    if (tid == 0) {
        double var = red[0] / 8388607.0;     // M-1, M = 2048*64*64, ddof=1
        double sd = sqrt(var);
        scale[b] = (float)(1.0 / (sd + 1e-5));
    }
}

// ---------------- gather + normalize + affine ----------------

__global__ __launch_bounds__(256) void k_out(const float* __restrict__ xy,
                                             const float* __restrict__ events,
                                             const int* __restrict__ idxws,
                                             const float* __restrict__ means,
                                             const float* __restrict__ scale,
                                             const float* __restrict__ alpha,
                                             const float* __restrict__ beta,
                                             float* __restrict__ outev) {
    __shared__ int sidx[NS];
    const int tid = threadIdx.x;
    const int b = blockIdx.x / S_;
    const int s = blockIdx.x % S_;
    if (tid < NS) sidx[tid] = idxws[(b * S_ + s) * NS + tid];
    __syncthreads();

    const int c = tid & 63;
    const float mean = means[(b * S_ + s) * CH + c];
    const float a  = alpha[c] * scale[b];
    const float bt = beta[c];
    float* orow = outev + (size_t)(b * S_ + s) * NS * CH;
    for (int k = (tid >> 6); k < NS; k += 4) {
        int i = sidx[k];
        float v = (c < D_) ? events[(b * N_ + i) * D_ + c]
                           : xy[(b * N_ + i) * 2 + (c - D_)];
        orow[k * CH + c] = a * (v - mean) + bt;
    }
}

// ---------------- launcher ----------------

extern "C" void kernel_launch(void* const* d_in, const int* in_sizes, int n_in,
                              void* d_out, int out_size, void* d_ws, size_t ws_size,
                              hipStream_t stream) {
    const float* xy     = (const float*)d_in[0];
    const float* events = (const float*)d_in[1];
    const int*   fps    = (const int*)d_in[2];
    const float* alpha  = (const float*)d_in[3];
    const float* beta   = (const float*)d_in[4];

    float* newxy = (float*)d_out;              // B*S*2 floats
    float* outev = (float*)d_out + B_ * S_ * 2;

    char* ws = (char*)d_ws;
    double* partials = (double*)ws;  ws += sizeof(double) * B_ * S_;        // 64 KB (8B aligned first)
    float4* pprep    = (float4*)ws;  ws += sizeof(float4) * B_ * N_;        // 1 MB
    float4* qprep    = (float4*)ws;  ws += sizeof(float4) * B_ * S_;        // 128 KB
    int*    idxws    = (int*)ws;     ws += sizeof(int) * B_ * S_ * NS;      // 2 MB
    float*  means    = (float*)ws;   ws += sizeof(float) * B_ * S_ * CH;    // 2 MB
    float*  scale    = (float*)ws;   ws += sizeof(float) * B_;              // 16 B

    k_prep_p<<<(B_ * N_ + 255) / 256, 256, 0, stream>>>(xy, pprep);
    k_prep_q<<<(B_ * S_ + 255) / 256, 256, 0, stream>>>(xy, fps, qprep, newxy);
    k_search<<<B_ * (S_ / 16), 32, 0, stream>>>(pprep, qprep, idxws);
    k_stats<<<B_ * S_, 64, 0, stream>>>(xy, events, idxws, means, partials);
    k_reduce<<<B_, 256, 0, stream>>>(partials, scale);
    k_out<<<B_ * S_, 256, 0, stream>>>(xy, events, idxws, means, scale, alpha, beta, outev);
}